// AttnDecoderRNN_18081812316456
// MI455X (gfx1250) — compile-verified
//
#include <hip/hip_runtime.h>
#include <hip/hip_bf16.h>
#include <math.h>

typedef __attribute__((ext_vector_type(2))) float v2f;
typedef __attribute__((ext_vector_type(8))) float v8f;

#define SEQ 30
#define H 64
#define VOCAB 128000

// ---------------------------------------------------------------------------
// Kernel 1: attention-applied vector (all-ones weights -> column sum) + GRU step
// 64 threads, one per hidden dim.
// ---------------------------------------------------------------------------
__global__ void gru_kernel(const float* __restrict__ input,
                           const float* __restrict__ hidden,
                           const float* __restrict__ W_ih,
                           const float* __restrict__ W_hh,
                           const float* __restrict__ b_ih,
                           const float* __restrict__ b_hh,
                           float* __restrict__ h_new_ws,
                           float* __restrict__ h_new_out) {
    __shared__ float xs[H];
    __shared__ float hs[H];
    const int d = threadIdx.x;  // 0..63

    // attn_weights == softmax over singleton dim == all ones  =>  x = sum_s input[s,0,:]
    float s = 0.0f;
    #pragma unroll
    for (int t = 0; t < SEQ; ++t) s += input[t * H + d];
    xs[d] = s;
    hs[d] = hidden[d];
    __syncthreads();

    float xr = b_ih[d], xz = b_ih[H + d], xn = b_ih[2 * H + d];
    float hr = b_hh[d], hz = b_hh[H + d], hn = b_hh[2 * H + d];
    #pragma unroll 4
    for (int j = 0; j < H; ++j) {
        const float xj = xs[j], hj = hs[j];
        xr = fmaf(W_ih[(0 * H + d) * H + j], xj, xr);
        xz = fmaf(W_ih[(1 * H + d) * H + j], xj, xz);
        xn = fmaf(W_ih[(2 * H + d) * H + j], xj, xn);
        hr = fmaf(W_hh[(0 * H + d) * H + j], hj, hr);
        hz = fmaf(W_hh[(1 * H + d) * H + j], hj, hz);
        hn = fmaf(W_hh[(2 * H + d) * H + j], hj, hn);
    }
    const float r = 1.0f / (1.0f + expf(-(xr + hr)));
    const float z = 1.0f / (1.0f + expf(-(xz + hz)));
    const float n = tanhf(xn + r * hn);
    const float h = (1.0f - z) * n + z * hs[d];
    h_new_ws[d]  = h;   // for the GEMV kernel
    h_new_out[d] = h;   // second tuple output
}

// ---------------------------------------------------------------------------
// Kernel 2: logits[v] = dot(W_out[v,:], h_new) + b_out[v] via V_WMMA_F32_16X16X4_F32
//
// Per wave32: A-tile = 16 vocab rows x 4 k (lane r<16 holds W[v0+r, k0+0..1],
// lane r+16 holds W[v0+r, k0+2..3] -> one b64 load/lane, tile streamed exactly
// once). B = h chunk broadcast across all 16 columns, so every column of D is
// the same dot product; after 16 accumulating WMMAs over H=64, lane 0 holds
// y[v0+0..7] and lane 16 holds y[v0+8..15] in the 8 accumulator VGPRs.
// 256 threads = 8 waves = 128 rows/block; 1000 blocks = 128000 rows.
// ---------------------------------------------------------------------------
__global__ void __launch_bounds__(256) gemv_wmma(const float* __restrict__ W,
                                                 const float* __restrict__ bias,
                                                 const float* __restrict__ hvec,
                                                 float* __restrict__ logits) {
    __shared__ float xs[H];
    if (threadIdx.x < H) xs[threadIdx.x] = hvec[threadIdx.x];
    __syncthreads();

    const int wave = threadIdx.x >> 5;
    const int lane = threadIdx.x & 31;
    const int half = lane >> 4;   // 0: K={0,1}, 1: K={2,3}
    const int r    = lane & 15;   // A-matrix row (vocab row within tile)
    const int v0   = blockIdx.x * 128 + wave * 16;

    const float* wrow = W + (size_t)(v0 + r) * H + 2 * half;
    const float* xrow = xs + 2 * half;

    v8f acc = {};
    #pragma unroll
    for (int k0 = 0; k0 < H; k0 += 4) {
        v2f a = *(const v2f*)(wrow + k0);          // W[v0+r, k0+2h .. k0+2h+1]
        v2f b;
        b.x = xrow[k0];                            // x[k0+2h]
        b.y = xrow[k0 + 1];                        // x[k0+2h+1]
        // D = A x B + C  (f32, 16x16x4); all lanes active (EXEC all-ones)
        acc = __builtin_amdgcn_wmma_f32_16x16x4_f32(
            /*neg_a=*/false, a, /*neg_b=*/false, b,
            /*c_mod=*/(short)0, acc, /*reuse_a=*/false, /*reuse_b=*/false);
    }

    // D VGPR i: lanes 0-15 -> (M=i, N=lane), lanes 16-31 -> (M=8+i).
    // Every N column is identical, so lanes with N==0 write the 16 results.
    if (r == 0) {
        const int mb = v0 + half * 8;
        #pragma unroll
        for (int i = 0; i < 8; ++i) logits[mb + i] = acc[i] + bias[mb + i];
    }
}

// ---------------------------------------------------------------------------
// Kernel 3: single-block deterministic max / log-sum-exp over 128000 logits
// (512 KB per pass, L2-resident; two passes << W_out streaming time).
// ---------------------------------------------------------------------------
__global__ void __launch_bounds__(1024) softmax_stats(const float* __restrict__ logits,
                                                      float* __restrict__ stats) {
    __shared__ float red[1024];
    const int tid = threadIdx.x;

    float m = -INFINITY;
    for (int i = tid; i < VOCAB; i += 1024) m = fmaxf(m, logits[i]);
    red[tid] = m;
    __syncthreads();
    for (int s = 512; s > 0; s >>= 1) {
        if (tid < s) red[tid] = fmaxf(red[tid], red[tid + s]);
        __syncthreads();
    }
    const float gmax = red[0];
    __syncthreads();

    float sum = 0.0f;
    for (int i = tid; i < VOCAB; i += 1024) sum += expf(logits[i] - gmax);
    red[tid] = sum;
    __syncthreads();
    for (int s = 512; s > 0; s >>= 1) {
        if (tid < s) red[tid] += red[tid + s];
        __syncthreads();
    }
    if (tid == 0) {
        stats[0] = gmax;
        stats[1] = logf(red[0]);
    }
}

// ---------------------------------------------------------------------------
// Kernel 4: out[v] = logit[v] - gmax - log(sum exp)
// ---------------------------------------------------------------------------
__global__ void __launch_bounds__(256) finalize(float* __restrict__ out,
                                                const float* __restrict__ stats) {
    const int i = blockIdx.x * 256 + threadIdx.x;
    if (i < VOCAB) out[i] = out[i] - stats[0] - stats[1];
}

extern "C" void kernel_launch(void* const* d_in, const int* in_sizes, int n_in,
                              void* d_out, int out_size, void* d_ws, size_t ws_size,
                              hipStream_t stream) {
    (void)in_sizes; (void)n_in; (void)out_size; (void)ws_size;
    const float* input  = (const float*)d_in[0];   // [30,1,64]
    const float* hidden = (const float*)d_in[1];   // [1,1,64]
    // d_in[2] W_attn, d_in[3] b_attn: dead code (softmax over singleton dim -> ones)
    const float* W_ih   = (const float*)d_in[4];   // [192,64]
    const float* W_hh   = (const float*)d_in[5];   // [192,64]
    const float* b_ih   = (const float*)d_in[6];   // [192]
    const float* b_hh   = (const float*)d_in[7];   // [192]
    const float* W_out  = (const float*)d_in[8];   // [128000,64]
    const float* b_out  = (const float*)d_in[9];   // [128000]
    // d_in[10] time_step: unused

    float* out = (float*)d_out;  // [0:128000] log-probs, [128000:128064] h_new
    float* ws  = (float*)d_ws;   // ws[0:64] h_new, ws[64:66] {gmax, log(sumexp)}

    gru_kernel<<<1, 64, 0, stream>>>(input, hidden, W_ih, W_hh, b_ih, b_hh,
                                     ws, out + VOCAB);
    gemv_wmma<<<VOCAB / 128, 256, 0, stream>>>(W_out, b_out, ws, out);
    softmax_stats<<<1, 1024, 0, stream>>>(out, ws + H);
    finalize<<<(VOCAB + 255) / 256, 256, 0, stream>>>(out, ws + H);
}